// AdaptiveTokenSampling_46686294507543
// MI455X (gfx1250) — compile-verified
//
#include <hip/hip_runtime.h>
#include <hip/hip_bf16.h>
#include <float.h>
#include <math.h>

typedef __attribute__((ext_vector_type(2))) float v2f;
typedef __attribute__((ext_vector_type(8))) float v8f;

#define EPSF 1e-6f
#define NTOK 256
#define BB 4
#define HH 16
#define NN 1024
#define DD 64
#define K1 257  // NTOK + 1 (prepended zero)

// ---------------------------------------------------------------------------
// Kernel 1: partial entropy per (b, h, 16-row tile) using V_WMMA_F32_16X16X4_F32.
// Accumulate Gram = V_tile * V_tile^T over K=64 (16 steps of K=4); the diagonal
// is sum-of-squares per row -> norm -> -norm*log(norm+1e-9). One wave per block
// so EXEC is all 1s (WMMA requirement). 32-bit A(16x4) layout: lanes 0-15 feed
// rows M=0..15 with K=(0,1), lanes 16-31 same rows with K=(2,3); B(4x16) with
// B=A^T has the identical per-lane contents, so we pass `a` for both operands.
// ---------------------------------------------------------------------------
__global__ __launch_bounds__(32) void k_entropy_partial(
    const float* __restrict__ value, float* __restrict__ partial) {
  const int t = blockIdx.x;            // tile 0..63 (rows 1+16t .. 1+16t+15)
  const int h = blockIdx.y, b = blockIdx.z;
  const int lane = threadIdx.x;        // 0..31 (wave32)
  const float* V = value + ((size_t)(b * HH + h)) * NN * DD;

  const int m = lane & 15;             // matrix row this lane feeds
  const int row = 1 + t * 16 + m;      // value row; row==1024 is out of range
  const int rclamp = (row > NN - 1) ? (NN - 1) : row;   // keep loads in-bounds
  const float vmul = (row > NN - 1) ? 0.0f : 1.0f;      // zero invalid row
  const int kofs = (lane >> 4) * 2;    // lanes 16-31 feed K+2, K+3

  const float* base = V + (size_t)rclamp * DD + kofs;   // 8B-aligned float2s
  v8f c = {};
#pragma unroll
  for (int kk = 0; kk < 16; ++kk) {
    float2 av = *(const float2*)(base + kk * 4);
    v2f a;
    a.x = av.x * vmul;
    a.y = av.y * vmul;
    // 8 args: (neg_a, A, neg_b, B, c_mod, C, reuse_a, reuse_b)
    c = __builtin_amdgcn_wmma_f32_16x16x4_f32(false, a, false, a, (short)0, c,
                                              false, false);
  }

  // Diagonal extraction. C/D layout: VGPR r, lanes 0-15 -> (M=r, N=lane);
  // lanes 16-31 -> (M=8+r, N=lane-16). So diag(m,m): m<8 at lane m, c[m];
  // m>=8 at lane m+16, c[m-8]; lanes 16-23 read c[lane-16] = (M=lane-8, N=lane-16).
  const int sel = lane & 7;
  float dv = c[0];
  dv = (sel == 1) ? c[1] : dv;
  dv = (sel == 2) ? c[2] : dv;
  dv = (sel == 3) ? c[3] : dv;
  dv = (sel == 4) ? c[4] : dv;
  dv = (sel == 5) ? c[5] : dv;
  dv = (sel == 6) ? c[6] : dv;
  dv = (sel == 7) ? c[7] : dv;

  const bool owns = (lane < 8) || (lane >= 16 && lane < 24);
  float term = 0.0f;
  if (owns) {
    const int myrow = 1 + t * 16 + ((lane < 8) ? lane : (lane - 8));
    float nrm = sqrtf(dv);
    term = -(nrm * logf(nrm + 1e-9f));
    if (myrow > NN - 1) term = 0.0f;
  }
  // wave32 butterfly reduction
#pragma unroll
  for (int off = 16; off > 0; off >>= 1) term += __shfl_xor(term, off, 32);
  if (lane == 0) partial[((b * HH + h) << 6) + t] = term;
}

// ---------------------------------------------------------------------------
// Kernel 2: per batch b (1024 threads): finish entropy[h], cls_score[i],
// block-reduce denominator, write masked pseudo-logits to ws.
// ---------------------------------------------------------------------------
__global__ __launch_bounds__(1024) void k_score(
    const float* __restrict__ attn, const unsigned char* __restrict__ mask,
    const float* __restrict__ partial, float* __restrict__ pl) {
  const int b = blockIdx.x;
  const int tid = threadIdx.x;
  __shared__ float ent[HH];
  __shared__ float red[1024];

  if (tid < HH) {
    const float* p = partial + ((b * HH + tid) << 6);
    float s = 0.0f;
#pragma unroll
    for (int j = 0; j < 64; ++j) s += p[j];
    ent[tid] = s;
  }
  __syncthreads();

  float cls = 0.0f;
  if (tid < NN - 1) {
#pragma unroll
    for (int h = 0; h < HH; ++h)
      cls += attn[((size_t)(b * HH + h)) * NN * NN + 1 + tid] * ent[h];
  }
  red[tid] = cls;
  __syncthreads();
  for (int s = 512; s > 0; s >>= 1) {
    if (tid < s) red[tid] += red[tid + s];
    __syncthreads();
  }
  const float total = red[0];

  if (tid < NN - 1) {
    float v = logf(cls / (total + EPSF) + EPSF);
    if (!mask[b * NN + 1 + tid]) v = -FLT_MAX * 0.5f;  // -finfo.max/2
    pl[b * NN + tid] = v;
  }
}

// ---------------------------------------------------------------------------
// Kernel 3: per (b, draw t): argmax_i( pl[i] + gumbel(u[b,t,i]) ), first-index
// tie-break to match jnp.argmax. sampled = argmax + 1.
// ---------------------------------------------------------------------------
__global__ __launch_bounds__(256) void k_argmax(
    const float* __restrict__ u, const float* __restrict__ pl,
    int* __restrict__ sampled) {
  const int t = blockIdx.x, b = blockIdx.y;
  const int tid = threadIdx.x;
  const float* urow = u + ((size_t)(b * NTOK + t)) * (NN - 1);
  const float* plrow = pl + b * NN;

  float best = -FLT_MAX;
  int bi = 0x7FFFFFFF;
  for (int i = tid; i < NN - 1; i += 256) {
    float g = -logf(-logf(urow[i] + EPSF) + EPSF);  // gumbel = -_log(-_log(u))
    float v = plrow[i] + g;
    if (v > best) { best = v; bi = i; }             // ascending i per thread
  }

  __shared__ float sv[256];
  __shared__ int si[256];
  sv[tid] = best;
  si[tid] = bi;
  __syncthreads();
  for (int s = 128; s > 0; s >>= 1) {
    if (tid < s) {
      float ov = sv[tid + s];
      int oi = si[tid + s];
      if (ov > sv[tid] || (ov == sv[tid] && oi < si[tid])) {
        sv[tid] = ov;
        si[tid] = oi;
      }
    }
    __syncthreads();
  }
  if (tid == 0) sampled[b * NTOK + t] = si[0] + 1;
}

// ---------------------------------------------------------------------------
// Kernel 4: per batch: bitonic sort 256 ids, dup -> sentinel(1024), re-sort,
// sentinel -> 0, prepend 0. Emits int uniq (for gather) + float mask/uniq out.
// ---------------------------------------------------------------------------
__device__ __forceinline__ void bitonic256(int* s, int tid) {
  for (int k = 2; k <= 256; k <<= 1) {
    for (int j = k >> 1; j > 0; j >>= 1) {
      int ixj = tid ^ j;
      if (ixj > tid) {
        int a = s[tid], c = s[ixj];
        bool up = (tid & k) == 0;
        if ((a > c) == up) { s[tid] = c; s[ixj] = a; }
      }
      __syncthreads();
    }
  }
}

__global__ __launch_bounds__(256) void k_uniq(
    const int* __restrict__ sampled, int* __restrict__ uniq,
    float* __restrict__ out_mask, float* __restrict__ out_uniq) {
  const int b = blockIdx.x, tid = threadIdx.x;
  __shared__ int s[256];
  s[tid] = sampled[b * NTOK + tid];
  __syncthreads();

  bitonic256(s, tid);                       // ascending sort

  int v = s[tid];
  int prev = (tid > 0) ? s[tid - 1] : -1;
  __syncthreads();
  s[tid] = (tid > 0 && v == prev) ? NN : v; // dup -> sentinel n=1024
  __syncthreads();

  bitonic256(s, tid);                       // sentinels float to the end

  int uv = (s[tid] == NN) ? 0 : s[tid];
  if (tid == 0) {
    uniq[b * K1] = 0;
    out_mask[b * K1] = 1.0f;                // mask[:,0] forced True
    out_uniq[b * K1] = 0.0f;
  }
  uniq[b * K1 + 1 + tid] = uv;
  out_mask[b * K1 + 1 + tid] = (uv != 0) ? 1.0f : 0.0f;
  out_uniq[b * K1 + 1 + tid] = (float)uv;
}

// ---------------------------------------------------------------------------
// Kernel 5: the HBM-bound part. Gather 257 rows (4 KB each, contiguous) per
// (b,h): one float4 (global_load_b128/global_store_b128) per lane.
// ~67 MB read + 67 MB write -> ~5.8 us at 23.3 TB/s.
// ---------------------------------------------------------------------------
__global__ __launch_bounds__(256) void k_gather(
    const float* __restrict__ attn, const int* __restrict__ uniq,
    float* __restrict__ out) {
  const int k = blockIdx.x, h = blockIdx.y, b = blockIdx.z;
  const int tid = threadIdx.x;
  const int row = uniq[b * K1 + k];
  const float4* src =
      (const float4*)(attn + (((size_t)(b * HH + h)) * NN + row) * NN);
  float4* dst = (float4*)(out + (((size_t)(b * HH + h)) * K1 + k) * NN);
  dst[tid] = src[tid];
}

// ---------------------------------------------------------------------------
extern "C" void kernel_launch(void* const* d_in, const int* in_sizes, int n_in,
                              void* d_out, int out_size, void* d_ws,
                              size_t ws_size, hipStream_t stream) {
  (void)in_sizes; (void)n_in; (void)out_size; (void)ws_size;
  const float* attn = (const float*)d_in[0];          // (4,16,1024,1024) f32
  const float* value = (const float*)d_in[1];         // (4,16,1024,64)   f32
  const float* u = (const float*)d_in[2];             // (4,256,1023)     f32
  const unsigned char* mask = (const unsigned char*)d_in[3];  // (4,1024) bool

  float* out = (float*)d_out;
  float* out_attn = out;                                    // 4*16*257*1024
  float* out_mask = out + (size_t)BB * HH * K1 * NN;        // 4*257
  float* out_uniq = out_mask + BB * K1;                     // 4*257

  // Workspace layout (all slots written before read each call; ~41 KB)
  float* partial = (float*)d_ws;                  // B*H*64 partial entropies
  float* pl = partial + BB * HH * 64;             // B*N pseudo-logits
  int* sampled = (int*)(pl + BB * NN);            // B*256 sampled ids
  int* uniq = sampled + BB * NTOK;                // B*257 gather indices

  k_entropy_partial<<<dim3(64, HH, BB), 32, 0, stream>>>(value, partial);
  k_score<<<dim3(BB), 1024, 0, stream>>>(attn, mask, partial, pl);
  k_argmax<<<dim3(NTOK, BB), 256, 0, stream>>>(u, pl, sampled);
  k_uniq<<<dim3(BB), 256, 0, stream>>>(sampled, uniq, out_mask, out_uniq);
  k_gather<<<dim3(K1, HH, BB), 256, 0, stream>>>(attn, uniq, out_attn);
}